// GAT_38517266710930
// MI455X (gfx1250) — compile-verified
//
#include <hip/hip_runtime.h>
#include <hip/hip_bf16.h>
#include <math.h>

typedef float v2f __attribute__((ext_vector_type(2)));
typedef float v8f __attribute__((ext_vector_type(8)));

#define NEG_SLOPE 0.2f
#define EPS_SM 1e-16f

// ---------------------------------------------------------------------------
// WMMA fp32 GEMM: C[M,N] = A[M,K] * B[K,N], row-major, M%16==0, N%16==0, K%4==0
// One 16x16 output tile per wave32; 4 waves per block.
// Uses V_WMMA_F32_16X16X4_F32 (exact fp32 -> matches fp32 reference).
// A-frag layout (ISA 7.12.2, 32-bit A 16x4): lane L: m=L%16, k = 2*(L/16)+v
// C/D layout: VGPR r: lanes 0-15 -> M=r, lanes 16-31 -> M=r+8, N = L%16
// ---------------------------------------------------------------------------
__global__ void wmma_gemm_f32(const float* __restrict__ A,
                              const float* __restrict__ B,
                              float* __restrict__ C,
                              int M, int K, int N) {
    const int tilesN = N >> 4;
    const int tiles  = (M >> 4) * tilesN;
    const int waveId = blockIdx.x * (blockDim.x >> 5) + (threadIdx.x >> 5);
    if (waveId >= tiles) return;                // wave-uniform exit (EXEC all-1 inside)
    const int tm = waveId / tilesN;
    const int tn = waveId % tilesN;

    const int lane = threadIdx.x & 31;
    const int hv   = lane >> 4;                 // 0 (lanes 0-15) / 1 (lanes 16-31)
    const int l16  = lane & 15;

    const float* Arow = A + (size_t)(tm * 16 + l16) * K;
    const float* Bcol = B + (tn * 16 + l16);

    v8f acc = {};
    for (int k = 0; k < K; k += 4) {
        const int kk = k + 2 * hv;
        v2f a, b;
        a.x = Arow[kk];
        a.y = Arow[kk + 1];
        b.x = Bcol[(size_t)kk * N];
        b.y = Bcol[(size_t)(kk + 1) * N];
        acc = __builtin_amdgcn_wmma_f32_16x16x4_f32(
            /*neg_a=*/false, a, /*neg_b=*/false, b,
            /*c_mod=*/(short)0, acc, /*reuse_a=*/false, /*reuse_b=*/false);
    }
#pragma unroll
    for (int r = 0; r < 8; ++r) {
        const int m = tm * 16 + r + 8 * hv;
        C[(size_t)m * N + tn * 16 + l16] = acc[r];
    }
}

// ---------------------------------------------------------------------------
// Per-node attention logits: as[n,h] = sum_c h[n,h,c]*att_s[h,c] (same for ad)
// ---------------------------------------------------------------------------
__global__ void node_logits_kernel(const float* __restrict__ h,
                                   const float* __restrict__ att_s,
                                   const float* __restrict__ att_d,
                                   float* __restrict__ as, float* __restrict__ ad,
                                   int NN, int H, int C) {
    const int i = blockIdx.x * blockDim.x + threadIdx.x;
    if (i >= NN * H) return;
    const int n  = i / H;
    const int hd = i % H;
    const float* row = h + ((size_t)n * H + hd) * C;
    const float* asv = att_s + hd * C;
    const float* adv = att_d + hd * C;
    float s = 0.f, d = 0.f;
    for (int c = 0; c < C; ++c) {
        const float v = row[c];
        s = fmaf(v, asv[c], s);
        d = fmaf(v, adv[c], d);
    }
    as[i] = s;
    ad[i] = d;
}

__device__ __forceinline__ void edge_ids(int e, const int* __restrict__ ei, int E,
                                         int& s, int& d) {
    if (e < E) { s = ei[e]; d = ei[E + e]; }
    else       { s = e - E; d = e - E; }       // synthesized self-loop
}

__device__ __forceinline__ float leaky(float x) {
    return x > 0.f ? x : NEG_SLOPE * x;
}

__device__ __forceinline__ void atomicMaxF(float* addr, float val) {
    int* a = (int*)addr;
    int old = __hip_atomic_load(a, __ATOMIC_RELAXED, __HIP_MEMORY_SCOPE_AGENT);
    while (val > __int_as_float(old)) {
        const int assumed = old;
        old = atomicCAS(a, assumed, __float_as_int(val));
        if (old == assumed) break;
    }
}

// Pass 1: segment max of leaky_relu logits per (dst, head)
__global__ void edge_max_kernel(const int* __restrict__ ei, int E, int NN, int H,
                                const float* __restrict__ as,
                                const float* __restrict__ ad,
                                float* __restrict__ m) {
    const int i = blockIdx.x * blockDim.x + threadIdx.x;
    if (i >= (E + NN) * H) return;
    const int e  = i / H;
    const int hd = i % H;
    int s, d; edge_ids(e, ei, E, s, d);
    const float x = leaky(as[s * H + hd] + ad[d * H + hd]);
    atomicMaxF(&m[d * H + hd], x);
}

// Pass 2: segment sum of exp(e - m) per (dst, head)
__global__ void edge_sum_kernel(const int* __restrict__ ei, int E, int NN, int H,
                                const float* __restrict__ as,
                                const float* __restrict__ ad,
                                const float* __restrict__ m,
                                float* __restrict__ denom) {
    const int i = blockIdx.x * blockDim.x + threadIdx.x;
    if (i >= (E + NN) * H) return;
    const int e  = i / H;
    const int hd = i % H;
    int s, d; edge_ids(e, ei, E, s, d);
    const float x = leaky(as[s * H + hd] + ad[d * H + hd]);
    atomicAdd(&denom[d * H + hd], expf(x - m[d * H + hd]));
}

// Pass 3: out[dst, h, c] += h[src, h, c] * alpha   (one block per edge, HC threads)
__global__ void edge_agg_kernel(const int* __restrict__ ei, int E, int NN,
                                int H, int C,
                                const float* __restrict__ feat,
                                const float* __restrict__ as,
                                const float* __restrict__ ad,
                                const float* __restrict__ m,
                                const float* __restrict__ denom,
                                float* __restrict__ out) {
    const int e = blockIdx.x;
    const int t = threadIdx.x;                  // t in [0, H*C)
    const int hd = t / C;
    int s, d; edge_ids(e, ei, E, s, d);
    const float x = leaky(as[s * H + hd] + ad[d * H + hd]);
    const float alpha = expf(x - m[d * H + hd]) / (denom[d * H + hd] + EPS_SM);
    atomicAdd(&out[(size_t)d * H * C + t], feat[(size_t)s * H * C + t] * alpha);
}

__global__ void fill_kernel(float* __restrict__ p, float v, int n) {
    const int i = blockIdx.x * blockDim.x + threadIdx.x;
    if (i < n) p[i] = v;
}

// h1 = elu(agg + b1), in place
__global__ void bias_elu_kernel(float* __restrict__ x, const float* __restrict__ b,
                                int NN, int F) {
    const int i = blockIdx.x * blockDim.x + threadIdx.x;
    if (i >= NN * F) return;
    const float v = x[i] + b[i % F];
    x[i] = v > 0.f ? v : (expf(v) - 1.f);
}

// out += b2, in place
__global__ void bias_add_kernel(float* __restrict__ x, const float* __restrict__ b,
                                int NN, int F) {
    const int i = blockIdx.x * blockDim.x + threadIdx.x;
    if (i >= NN * F) return;
    x[i] += b[i % F];
}

// ---------------------------------------------------------------------------

static inline int ceil_div(long long a, long long b) { return (int)((a + b - 1) / b); }

extern "C" void kernel_launch(void* const* d_in, const int* in_sizes, int n_in,
                              void* d_out, int out_size, void* d_ws, size_t ws_size,
                              hipStream_t stream) {
    const float* x        = (const float*)d_in[0];
    const int*   ei       = (const int*)d_in[1];   // [2,E]: src row then dst row
    const float* W1       = (const float*)d_in[2];
    const float* att_src1 = (const float*)d_in[3];
    const float* att_dst1 = (const float*)d_in[4];
    const float* b1       = (const float*)d_in[5];
    const float* W2       = (const float*)d_in[6];
    const float* att_src2 = (const float*)d_in[7];
    const float* att_dst2 = (const float*)d_in[8];
    const float* b2       = (const float*)d_in[9];
    float* out = (float*)d_out;

    const int F1   = in_sizes[5];               // 192 = HEADS*HID
    const int OUT  = in_sizes[9];               // 64
    const int IN   = in_sizes[2] / F1;          // 128
    const int NN   = in_sizes[0] / IN;          // 50000
    const int E    = in_sizes[1] / 2;           // 800000
    const int HID  = OUT;                       // 64
    const int H    = F1 / HID;                  // 3
    const int ET   = E + NN;                    // edges incl. self loops

    // ---- workspace carve-out ----
    char* w = (char*)d_ws;
    float* hbuf = (float*)w; w += (size_t)NN * F1 * sizeof(float);   // h, later h2pre
    float* agg1 = (float*)w; w += (size_t)NN * F1 * sizeof(float);   // agg, later h1
    float* as1  = (float*)w; w += (size_t)NN * H * sizeof(float);
    float* ad1  = (float*)w; w += (size_t)NN * H * sizeof(float);
    float* m1   = (float*)w; w += (size_t)NN * H * sizeof(float);
    float* dn1  = (float*)w; w += (size_t)NN * H * sizeof(float);
    float* as2  = (float*)w; w += (size_t)NN * sizeof(float);
    float* ad2  = (float*)w; w += (size_t)NN * sizeof(float);
    float* m2   = (float*)w; w += (size_t)NN * sizeof(float);
    float* dn2  = (float*)w; w += (size_t)NN * sizeof(float);

    const int TB = 256;

    // ================= Layer 1 =================
    // h = x @ W1
    {
        const int tiles  = (NN / 16) * (F1 / 16);
        const int blocks = ceil_div(tiles, 4);   // 4 waves/block
        wmma_gemm_f32<<<blocks, 128, 0, stream>>>(x, W1, hbuf, NN, IN, F1);
    }
    // per-node logits
    node_logits_kernel<<<ceil_div((long long)NN * H, TB), TB, 0, stream>>>(
        hbuf, att_src1, att_dst1, as1, ad1, NN, H, HID);
    // init m = -inf, denom = 0, agg = 0
    fill_kernel<<<ceil_div((long long)NN * H, TB), TB, 0, stream>>>(m1, -INFINITY, NN * H);
    fill_kernel<<<ceil_div((long long)NN * H, TB), TB, 0, stream>>>(dn1, 0.f, NN * H);
    fill_kernel<<<ceil_div((long long)NN * F1, TB), TB, 0, stream>>>(agg1, 0.f, NN * F1);
    // softmax passes + aggregate
    edge_max_kernel<<<ceil_div((long long)ET * H, TB), TB, 0, stream>>>(
        ei, E, NN, H, as1, ad1, m1);
    edge_sum_kernel<<<ceil_div((long long)ET * H, TB), TB, 0, stream>>>(
        ei, E, NN, H, as1, ad1, m1, dn1);
    edge_agg_kernel<<<ET, H * HID, 0, stream>>>(
        ei, E, NN, H, HID, hbuf, as1, ad1, m1, dn1, agg1);
    // h1 = elu(agg + b1) in place
    bias_elu_kernel<<<ceil_div((long long)NN * F1, TB), TB, 0, stream>>>(agg1, b1, NN, F1);

    // ================= Layer 2 =================
    // h2pre = h1 @ W2 (reuse hbuf)
    {
        const int tiles  = (NN / 16) * (OUT / 16);
        const int blocks = ceil_div(tiles, 4);
        wmma_gemm_f32<<<blocks, 128, 0, stream>>>(agg1, W2, hbuf, NN, F1, OUT);
    }
    node_logits_kernel<<<ceil_div((long long)NN, TB), TB, 0, stream>>>(
        hbuf, att_src2, att_dst2, as2, ad2, NN, 1, OUT);
    fill_kernel<<<ceil_div((long long)NN, TB), TB, 0, stream>>>(m2, -INFINITY, NN);
    fill_kernel<<<ceil_div((long long)NN, TB), TB, 0, stream>>>(dn2, 0.f, NN);
    fill_kernel<<<ceil_div((long long)NN * OUT, TB), TB, 0, stream>>>(out, 0.f, NN * OUT);
    edge_max_kernel<<<ceil_div((long long)ET, TB), TB, 0, stream>>>(
        ei, E, NN, 1, as2, ad2, m2);
    edge_sum_kernel<<<ceil_div((long long)ET, TB), TB, 0, stream>>>(
        ei, E, NN, 1, as2, ad2, m2, dn2);
    edge_agg_kernel<<<ET, OUT, 0, stream>>>(
        ei, E, NN, 1, OUT, hbuf, as2, ad2, m2, dn2, out);
    bias_add_kernel<<<ceil_div((long long)NN * OUT, TB), TB, 0, stream>>>(out, b2, NN, OUT);
}